// VoxelHuman_80333068304927
// MI455X (gfx1250) — compile-verified
//
#include <hip/hip_runtime.h>
#include <hip/hip_bf16.h>
#include <math.h>

typedef __attribute__((ext_vector_type(2))) float v2f;
typedef __attribute__((ext_vector_type(8))) float v8f;

#define HW   16384     // H*W rays
#define S_   64        // samples per ray
#define C_   132       // channels in raw (3 rgb + 1 sdf + 128 feat)
#define F_   128       // feature channels
#define INF_ 1e10f

// Output layout (NCHW, concatenated): rgb_map[3][HW], feature_map[128][HW],
// xyz[3][HW], mask[1][HW]  -> 135*16384 floats total.

__global__ __launch_bounds__(32)
void voxel_human_kernel(const float* __restrict__ raw,
                        const float* __restrict__ z_vals,
                        const float* __restrict__ rays_d,
                        const float* __restrict__ pts,
                        const float* __restrict__ beta_p,
                        float* __restrict__ out)
{
    // Weight pairs are read as 8-byte ds_load_b64 -> keep 16B alignment.
    __shared__ __align__(16) float wLDS[16 * S_];   // weights [ray-in-group][s]
    __shared__ __align__(16) float fLDS[F_ * 16];   // staging [feat][ray-in-group]

    const int lane = threadIdx.x;      // 0..31 (wave32)
    const int g    = blockIdx.x;       // 0..1023 ray groups
    const int ray0 = g * 16;
    const float beta = beta_p[0];

    // ---------------- Phase 1: weights, rgb, xyz, mask (lanes 0..15) -------
    if (lane < 16) {
        const int ray = ray0 + lane;
        const float* zr = z_vals + (size_t)ray * S_;
        const float* rd = rays_d + (size_t)ray * 3;
        const float rdn = sqrtf(rd[0]*rd[0] + rd[1]*rd[1] + rd[2]*rd[2]);

        float T = 1.0f, wsum = 0.0f;
        for (int s = 0; s < S_; ++s) {
            float dz    = (s < S_ - 1) ? (zr[s + 1] - zr[s]) : INF_;
            float dist  = dz * rdn;
            float sdf   = raw[((size_t)ray * S_ + s) * C_ + 3];
            // sigmoid(-sdf/beta)/beta
            float sig   = (1.0f / (1.0f + __expf(sdf / beta))) / beta;
            float alpha = 1.0f - __expf(-sig * dist);
            float w     = alpha * T;
            T *= (1.0f - alpha + 1e-10f);
            if (s < S_ - 1) wsum += w;
            wLDS[lane * S_ + s] = w;
        }
        const float w_last = 1.0f - wsum;            // overwrite last weight
        wLDS[lane * S_ + (S_ - 1)] = w_last;

        float r0 = 0.f, r1 = 0.f, r2 = 0.f, x0 = 0.f, x1 = 0.f, x2 = 0.f;
        for (int s = 0; s < S_; ++s) {
            float w = wLDS[lane * S_ + s];
            const float* rp_ = raw + ((size_t)ray * S_ + s) * C_;
            r0 += w * (1.0f / (1.0f + __expf(-rp_[0])));
            r1 += w * (1.0f / (1.0f + __expf(-rp_[1])));
            r2 += w * (1.0f / (1.0f + __expf(-rp_[2])));
            const float* pp = pts + ((size_t)ray * S_ + s) * 3;
            x0 += w * pp[0]; x1 += w * pp[1]; x2 += w * pp[2];
        }
        out[0 * HW + ray] = -1.0f + 2.0f * r0;
        out[1 * HW + ray] = -1.0f + 2.0f * r1;
        out[2 * HW + ray] = -1.0f + 2.0f * r2;
        out[(size_t)(3 + F_ + 0) * HW + ray] = x0;
        out[(size_t)(3 + F_ + 1) * HW + ray] = x1;
        out[(size_t)(3 + F_ + 2) * HW + ray] = x2;
        out[(size_t)(3 + F_ + 3) * HW + ray] = w_last;   // mask
    }
    __syncthreads();

    // ---------------- Phase 2: feature_map via V_WMMA_F32_16X16X4_F32 ------
    // D[m][n] = sum over K=(ray',s) of A[m][K]*B[K][n], where A holds the 4
    // weights of ray' only in row m==ray' (block-diagonal), B is the 4x16
    // feature slab of ray'. 8 N-tiles cover 128 features.
    v8f acc[8] = {};

    const int n    = lane & 15;          // N index (feature within tile) & M index
    const int kofs = (lane >> 4) << 1;   // lanes 0-15: K0/K1, lanes 16-31: K2/K3

    for (int rp = 0; rp < 16; ++rp) {
        // Weight pairs for this source ray, viewed as 2-float vectors.
        const v2f* wrow2 = reinterpret_cast<const v2f*>(&wLDS[rp * S_]);
        const size_t rbase = ((size_t)(ray0 + rp)) * S_;
        const bool mine = (n == rp);
        for (int sb = 0; sb < S_; sb += 4) {
            // Unconditional ds_load_b64, then register select (no exec masking).
            v2f wpair = wrow2[(sb + kofs) >> 1];
            v2f A;
            A.x = mine ? wpair.x : 0.0f;
            A.y = mine ? wpair.y : 0.0f;
            const float* fb = raw + (rbase + sb + kofs) * C_ + 4;
            #pragma unroll
            for (int t = 0; t < 8; ++t) {
                v2f B;
                B.x = fb[t * 16 + n];        // sample sb+kofs
                B.y = fb[C_ + t * 16 + n];   // sample sb+kofs+1
                acc[t] = __builtin_amdgcn_wmma_f32_16x16x4_f32(
                    false, A, false, B, (short)0, acc[t], false, false);
            }
        }
    }

    // ---------------- Phase 3: stage D tiles, coalesced NCHW store ----------
    // D layout: VGPR j -> M=j (lanes 0-15) / M=j+8 (lanes 16-31), N=lane&15.
    const int half = lane >> 4;
    #pragma unroll
    for (int t = 0; t < 8; ++t) {
        #pragma unroll
        for (int j = 0; j < 8; ++j) {
            fLDS[(t * 16 + n) * 16 + (j + half * 8)] = acc[t][j];
        }
    }
    __syncthreads();

    float* fout = out + (size_t)3 * HW;
    for (int i = lane; i < F_ * 16; i += 32) {
        int f = i >> 4, mm = i & 15;
        fout[(size_t)f * HW + ray0 + mm] = fLDS[i];
    }
}

extern "C" void kernel_launch(void* const* d_in, const int* in_sizes, int n_in,
                              void* d_out, int out_size, void* d_ws, size_t ws_size,
                              hipStream_t stream) {
    (void)in_sizes; (void)n_in; (void)out_size; (void)d_ws; (void)ws_size;
    const float* raw    = (const float*)d_in[0];
    const float* z_vals = (const float*)d_in[1];
    const float* rays_d = (const float*)d_in[2];
    const float* pts    = (const float*)d_in[3];
    const float* beta   = (const float*)d_in[4];
    float* out = (float*)d_out;
    voxel_human_kernel<<<HW / 16, 32, 0, stream>>>(raw, z_vals, rays_d, pts, beta, out);
}